// SessionGraph_49091476193716
// MI455X (gfx1250) — compile-verified
//
#include <hip/hip_runtime.h>
#include <hip/hip_bf16.h>

#define B_ 512
#define N_ 64
#define E_ 64
#define D_ 256
#define ALPHA_ 0.2f
#define NEGV -9.0e15f
#define LDX 257   // padded row stride for [64][256] LDS tiles (breaks 64-bank conflicts)
#define LDA 65    // padded row stride for [64][64] attention tiles

typedef float v2f __attribute__((ext_vector_type(2)));
typedef float v8f __attribute__((ext_vector_type(8)));

// ---------------------------------------------------------------------------
// M=64 x N=256 GEMM, C = A(64xK) * B(KxN), fp32 WMMA 16x16x4, 8 waves/block.
// A: LDS (lda-padded). B: LDS or global, row-major ldb. C: LDS, ldc-padded.
// Fragment layouts per CDNA5 ISA 7.12.2:
//   A 16x4 : lane 0-15 -> M=lane, VGPR j -> K = 2*(lane>=16) + j
//   B 4x16 : lane 0-15 -> N=lane, VGPR j -> K = 2*(lane>=16) + j
//   C 16x16: VGPR r -> M = r + 8*(lane>=16), N = lane&15
// ---------------------------------------------------------------------------
__device__ __forceinline__ void gemm64xN(const float* __restrict__ A, int lda,
                                         const float* __restrict__ Bm, int ldb,
                                         float* __restrict__ C, int ldc,
                                         int K, int tid)
{
  const int lane = tid & 31;
  const int l15  = lane & 15;
  const int half = lane >> 4;
  // 64 output tiles: 4 (M) x 16 (N); each of 8 waves takes 8 tiles
  for (int t = (tid >> 5); t < 64; t += 8) {
    const int m0 = (t & 3) << 4;
    const int n0 = (t >> 2) << 4;
    v8f acc = {0.f, 0.f, 0.f, 0.f, 0.f, 0.f, 0.f, 0.f};
    const float* ap = A  + (m0 + l15) * lda + (half << 1);
    const float* bp = Bm + (half << 1) * ldb + n0 + l15;
    for (int k = 0; k < K; k += 4) {
      v2f av, bv;
      av.x = ap[k];
      av.y = ap[k + 1];
      bv.x = bp[k * ldb];
      bv.y = bp[(k + 1) * ldb];
      acc = __builtin_amdgcn_wmma_f32_16x16x4_f32(false, av, false, bv,
                                                  (short)0, acc, false, false);
    }
    float* cp = C + (m0 + (half << 3)) * ldc + n0 + l15;
#pragma unroll
    for (int r = 0; r < 8; ++r) cp[r * ldc] = acc[r];
  }
}

// ---------------------------------------------------------------------------
// Embedding gathers: one block per (b,n) row, 256 threads = D floats.
// ---------------------------------------------------------------------------
__global__ void sg_gather(const int* __restrict__ idx,
                          const float* __restrict__ emb1,
                          const float* __restrict__ emb2,
                          float* __restrict__ x0,
                          float* __restrict__ nodes2)
{
  const int row = blockIdx.x;
  const int t = threadIdx.x;
  const int node = idx[row];
  x0[(size_t)row * D_ + t]     = emb1[(size_t)node * D_ + t];
  nodes2[(size_t)row * D_ + t] = emb2[(size_t)node * D_ + t];
}

// ---------------------------------------------------------------------------
// Fold attention vectors through the weight matrices (exact algebra):
//  u1 = w2 @ a[D:2D]   (node->edge score),  u2 = w2 @ a2[0:D]
//  v3 = w3 @ a2[D:2D]  (edge score)         c0 = wc . a[0:D]
// blocks 0..2 -> layer1 vectors, 3..5 -> layer2; vecs[1536/1537] = scalars.
// ---------------------------------------------------------------------------
__global__ void sg_prep(const float* __restrict__ w2_1, const float* __restrict__ w3_1,
                        const float* __restrict__ a_1,  const float* __restrict__ a2_1,
                        const float* __restrict__ wc_1,
                        const float* __restrict__ w2_2, const float* __restrict__ w3_2,
                        const float* __restrict__ a_2,  const float* __restrict__ a2_2,
                        const float* __restrict__ wc_2,
                        float* __restrict__ vecs)
{
  const int blk = blockIdx.x;
  const int t = threadIdx.x;
  const float* W = nullptr;
  const float* v = nullptr;
  switch (blk) {
    case 0: W = w2_1; v = a_1 + D_;  break;
    case 1: W = w2_1; v = a2_1;      break;
    case 2: W = w3_1; v = a2_1 + D_; break;
    case 3: W = w2_2; v = a_2 + D_;  break;
    case 4: W = w2_2; v = a2_2;      break;
    case 5: W = w3_2; v = a2_2 + D_; break;
  }
  float s = 0.f;
  for (int j = 0; j < D_; ++j) s += W[t * D_ + j] * v[j];
  vecs[blk * D_ + t] = s;
  if (blk == 0 && t == 0) {
    float c = 0.f;
    for (int j = 0; j < D_; ++j) c += wc_1[j] * a_1[j];
    vecs[1536] = c;
  }
  if (blk == 0 && t == 1) {
    float c = 0.f;
    for (int j = 0; j < D_; ++j) c += wc_2[j] * a_2[j];
    vecs[1537] = c;
  }
}

// ---------------------------------------------------------------------------
// One full HyperGAT layer per workgroup (one batch element each).
// 256 threads (8 wave32). Dynamic LDS:
//   sX  [64][LDX]  input x, later reused as node output
//   sE  [64][LDX]  edge features
//   sAtt[64][LDA]  att_edge, later reused as att_node
//   sS1/sSxn/sSe [64] score vectors
//   sXW [64][LDX]  x @ w (HAS_W only)
// ---------------------------------------------------------------------------
template <bool HAS_W>
__global__ void hgat_layer(const float* __restrict__ xin,
                           const float* resid,
                           const int* __restrict__ HT,
                           const float* __restrict__ w,
                           const float* __restrict__ u1,
                           const float* __restrict__ u2,
                           const float* __restrict__ v3,
                           const float* __restrict__ c0p,
                           float* out0,
                           float* out1)
{
  extern __shared__ float smem[];
  float* sX   = smem;
  float* sE   = sX + N_ * LDX;
  float* sAtt = sE + N_ * LDX;
  float* sS1  = sAtt + E_ * LDA;
  float* sSxn = sS1 + N_;
  float* sSe  = sSxn + N_;
  float* sXW  = sSe + E_;   // valid only when HAS_W

  const int b = blockIdx.x;
  const int tid = threadIdx.x;
  const float* xg = xin + (size_t)b * N_ * D_;
  const float* rg = resid + (size_t)b * N_ * D_;
  const int* htb = HT + (size_t)b * E_ * N_;

  // ---- load x into LDS (coalesced) ----
  for (int i = tid; i < N_ * D_; i += 256) {
    const int r = i >> 8, c = i & 255;
    sX[r * LDX + c] = xg[i];
  }
  __syncthreads();

  // ---- node scores: s1[n] = leaky(c0 + x[n].u1), sxn[n] = x[n].u2 ----
  if (tid < N_) {
    const float c0 = c0p[0];
    float d1 = 0.f, d2 = 0.f;
    const float* row = sX + tid * LDX;
    for (int k = 0; k < D_; ++k) {
      const float xv = row[k];
      d1 += xv * u1[k];
      d2 += xv * u2[k];
    }
    const float s = c0 + d1;
    sS1[tid] = s > 0.f ? s : ALPHA_ * s;
    sSxn[tid] = d2;
  }

  // ---- transfer GEMM (layer 2): xw = x @ w ----
  if (HAS_W) gemm64xN(sX, LDX, w, D_, sXW, LDX, D_, tid);
  __syncthreads();

  // ---- att_edge[e][n] = softmax_n(mask ? s1[n] : NEG) ----
  if (tid < E_) {
    const int e = tid;
    const int* hrow = htb + e * N_;
    unsigned mlo = 0u, mhi = 0u;
    float mx = -3.4e38f;
    for (int n = 0; n < N_; ++n) {
      const bool on = hrow[n] > 0;
      if (on) { if (n < 32) mlo |= 1u << n; else mhi |= 1u << (n - 32); }
      const float s = on ? sS1[n] : NEGV;
      mx = fmaxf(mx, s);
    }
    float sum = 0.f;
    for (int n = 0; n < N_; ++n) {
      const bool on = n < 32 ? ((mlo >> n) & 1u) : ((mhi >> (n - 32)) & 1u);
      const float s = on ? sS1[n] : NEGV;
      const float ez = __expf(s - mx);
      sAtt[e * LDA + n] = ez;
      sum += ez;
    }
    const float inv = 1.f / sum;
    for (int n = 0; n < N_; ++n) sAtt[e * LDA + n] *= inv;
  }
  __syncthreads();

  // ---- edge = att_edge @ xw ----
  gemm64xN(sAtt, LDA, HAS_W ? sXW : sX, LDX, sE, LDX, E_, tid);
  __syncthreads();

  // ---- se[e] = edge[e] . v3 ----
  if (tid < E_) {
    float d = 0.f;
    const float* row = sE + tid * LDX;
    for (int k = 0; k < D_; ++k) d += row[k] * v3[k];
    sSe[tid] = d;
  }
  __syncthreads();

  // ---- att_node[n][e] = softmax_e(mask[e][n] ? leaky(sxn[n]+se[e]) : NEG) ----
  if (tid < N_) {
    const int n = tid;
    const int* hcol = htb + n;          // stride N_ over e
    const float xn = sSxn[n];
    unsigned mlo = 0u, mhi = 0u;
    float mx = -3.4e38f;
    for (int e = 0; e < E_; ++e) {
      const bool on = hcol[e * N_] > 0;
      if (on) { if (e < 32) mlo |= 1u << e; else mhi |= 1u << (e - 32); }
      float t = xn + sSe[e];
      t = t > 0.f ? t : ALPHA_ * t;
      const float s = on ? t : NEGV;
      mx = fmaxf(mx, s);
    }
    float sum = 0.f;
    for (int e = 0; e < E_; ++e) {
      const bool on = e < 32 ? ((mlo >> e) & 1u) : ((mhi >> (e - 32)) & 1u);
      float t = xn + sSe[e];
      t = t > 0.f ? t : ALPHA_ * t;
      const float s = on ? t : NEGV;
      const float ez = __expf(s - mx);
      sAtt[n * LDA + e] = ez;
      sum += ez;
    }
    const float inv = 1.f / sum;
    for (int e = 0; e < E_; ++e) sAtt[n * LDA + e] *= inv;
  }
  __syncthreads();

  // ---- node = att_node @ edge  (into sX, x no longer needed) ----
  gemm64xN(sAtt, LDA, sE, LDX, sX, LDX, E_, tid);
  __syncthreads();

  // ---- epilogue: out = node + residual ----
  float* o0 = out0 + (size_t)b * N_ * D_;
  float* o1 = out1 ? out1 + (size_t)b * N_ * D_ : nullptr;
  for (int i = tid; i < N_ * D_; i += 256) {
    const int r = i >> 8, c = i & 255;
    const float v = sX[r * LDX + c] + rg[i];
    o0[i] = v;
    if (o1) o1[i] = v;
  }
}

// ---------------------------------------------------------------------------
extern "C" void kernel_launch(void* const* d_in, const int* in_sizes, int n_in,
                              void* d_out, int out_size, void* d_ws, size_t ws_size,
                              hipStream_t stream) {
  (void)in_sizes; (void)n_in; (void)out_size; (void)ws_size;
  const int*   inputs = (const int*)d_in[0];
  const int*   HT     = (const int*)d_in[1];
  const float* emb_w  = (const float*)d_in[4];
  const float* emb2_w = (const float*)d_in[5];
  const float* g1_w2  = (const float*)d_in[6];
  const float* g1_w3  = (const float*)d_in[7];
  const float* g1_a   = (const float*)d_in[8];
  const float* g1_a2  = (const float*)d_in[9];
  const float* g1_wc  = (const float*)d_in[10];
  const float* g2_w   = (const float*)d_in[11];
  const float* g2_w2  = (const float*)d_in[12];
  const float* g2_w3  = (const float*)d_in[13];
  const float* g2_a   = (const float*)d_in[14];
  const float* g2_a2  = (const float*)d_in[15];
  const float* g2_wc  = (const float*)d_in[16];

  float* out = (float*)d_out;
  const size_t seg = (size_t)B_ * N_ * D_;
  float* seg0 = out;            // final x (and layer-1 result ping buffer)
  float* seg1 = out + seg;      // x0 residual storage, then final x copy
  float* seg2 = out + 2 * seg;  // nodes2
  float* vecs = (float*)d_ws;   // 6*256 vectors + 2 scalars

  sg_gather<<<B_ * N_, 256, 0, stream>>>(inputs, emb_w, emb2_w, seg1, seg2);
  sg_prep<<<6, 256, 0, stream>>>(g1_w2, g1_w3, g1_a, g1_a2, g1_wc,
                                 g2_w2, g2_w3, g2_a, g2_a2, g2_wc, vecs);

  const size_t lds1 = (size_t)(2 * N_ * LDX + E_ * LDA + 3 * N_) * sizeof(float);
  const size_t lds2 = lds1 + (size_t)(N_ * LDX) * sizeof(float);

  // layer 1 (no transfer): x1 = layer(x0) + x0   -> seg0
  hgat_layer<false><<<B_, 256, lds1, stream>>>(seg1, seg1, HT, nullptr,
      vecs + 0, vecs + 256, vecs + 512, vecs + 1536, seg0, nullptr);
  // layer 2 (transfer w): x2 = layer(x1) + x0    -> seg0 and seg1
  hgat_layer<true><<<B_, 256, lds2, stream>>>(seg0, seg1, HT, g2_w,
      vecs + 768, vecs + 1024, vecs + 1280, vecs + 1537, seg0, seg1);
}